// Block_21955872817714
// MI455X (gfx1250) — compile-verified
//
#include <hip/hip_runtime.h>
#include <hip/hip_bf16.h>

// Problem constants (from the reference)
#define BB   8
#define CC   64
#define NN   3136          // 56*56
#define KK   16
#define OUTC 64
#define CNT  (BB * NN)     // 25088 points total
#define RT   196           // NN / 16 row tiles per batch
#define BN_EPS 1e-5f

typedef float v2f __attribute__((ext_vector_type(2)));
typedef float v8f __attribute__((ext_vector_type(8)));

// Full-precision CDNA5 matrix op: D(16x16,f32) = A(16x4,f32) * B(4x16,f32) + C
__device__ __forceinline__ v8f wmma_f32_k4(v2f a, v2f b, v8f c) {
    return __builtin_amdgcn_wmma_f32_16x16x4_f32(
        /*neg_a=*/false, a, /*neg_b=*/false, b,
        /*c_mod=*/(short)0, c, /*reuse_a=*/false, /*reuse_b=*/false);
}

// ---------------------------------------------------------------------------
// Utility: zero a small float buffer (BN stats accumulators) each launch.
// ---------------------------------------------------------------------------
__global__ void k_zero(float* __restrict__ p, int n) {
    int t = blockIdx.x * 256 + threadIdx.x;
    if (t < n) p[t] = 0.f;
}

// ---------------------------------------------------------------------------
// colnorm[b*C+c] = max(||x[b,c,:]||_2, 1e-12)   (F.normalize over points axis)
// ---------------------------------------------------------------------------
__global__ void k_colnorm(const float* __restrict__ x, float* __restrict__ colnorm) {
    __shared__ float red[256];
    const int bc = blockIdx.x;                 // b*C + c
    const float* col = x + (size_t)bc * NN;
    float s = 0.f;
    for (int n = threadIdx.x; n < NN; n += 256) { float v = col[n]; s += v * v; }
    red[threadIdx.x] = s;
    __syncthreads();
    for (int off = 128; off > 0; off >>= 1) {
        if (threadIdx.x < off) red[threadIdx.x] += red[threadIdx.x + off];
        __syncthreads();
    }
    if (threadIdx.x == 0) colnorm[bc] = fmaxf(sqrtf(red[0]), 1e-12f);
}

// ---------------------------------------------------------------------------
// pts[b,n,c] = x[b,c,n];  pnorm = pts / colnorm   ([B,N,C] row-major layout)
// ---------------------------------------------------------------------------
__global__ void k_transpose(const float* __restrict__ x, const float* __restrict__ colnorm,
                            float* __restrict__ pts, float* __restrict__ pnorm) {
    int t = blockIdx.x * 256 + threadIdx.x;    // (b*N + n)*C + c
    if (t >= BB * NN * CC) return;
    int c = t & 63;
    int n = (t >> 6) % NN;
    int b = t / (NN * CC);
    float v = x[((size_t)b * CC + c) * NN + n];
    pts[t]   = v;
    pnorm[t] = v / colnorm[b * CC + c];
}

// ---------------------------------------------------------------------------
// sqn[b*N+n] = sum_c pnorm^2
// ---------------------------------------------------------------------------
__global__ void k_sqnorm(const float* __restrict__ pnorm, float* __restrict__ sqn) {
    int t = blockIdx.x * 256 + threadIdx.x;
    if (t >= CNT) return;
    const float4* r = (const float4*)(pnorm + (size_t)t * CC);
    float s = 0.f;
#pragma unroll
    for (int i = 0; i < 16; ++i) { float4 v = r[i]; s += v.x*v.x + v.y*v.y + v.z*v.z + v.w*v.w; }
    sqn[t] = s;
}

// ---------------------------------------------------------------------------
// kNN: fused pairwise-distance GEMM (v_wmma_f32_16x16x4_f32) + top-16 select.
// One wave per 16-row tile, software-pipelined over column tiles with
// ping-pong B-operand register buffers (loads one tile ahead).
// ---------------------------------------------------------------------------
struct WaveScratch {
    float dist[16][17];    // one 16x16 distance tile (padded)
    float md[16][32];      // merge: per-row 2x16 candidate dists
    int   mi[16][32];      // merge: per-row 2x16 candidate indices
};

// Batched 16x global_load_b64 of one B tile (fp32 16x4 operand layout).
__device__ __forceinline__ void load_btile(v2f* bt, const float* br, int lh) {
#pragma unroll
    for (int s = 0; s < 16; ++s) {
        int k0 = 4 * s + (lh << 1);
        bt[s][0] = br[k0];
        bt[s][1] = br[k0 + 1];
    }
}

__device__ __forceinline__ void knn_tile(const v2f* __restrict__ a, const v2f* __restrict__ bt,
                                         int m0, const float* __restrict__ SQ,
                                         const float* __restrict__ rsq, WaveScratch* S,
                                         int l16, int lh,
                                         float* bd, int* bi, float& bmax, int& bpos) {
    v8f acc = {0.f, 0.f, 0.f, 0.f, 0.f, 0.f, 0.f, 0.f};
#pragma unroll
    for (int s = 0; s < 16; ++s) acc = wmma_f32_k4(a[s], bt[s], acc);
    const float csq = SQ[m0 + l16];
    // dist = |p_r|^2 - 2 <p_r,p_c> + |p_c|^2 ; stage 16x16 tile to LDS
#pragma unroll
    for (int j = 0; j < 8; ++j)
        S->dist[lh * 8 + j][l16] = rsq[j] - 2.0f * acc[j] + csq;
    __builtin_amdgcn_wave_barrier();
    // lane owns row l16, column half lh; replace-max top-16 (rare after warmup)
    const int c0 = lh << 3;
#pragma unroll
    for (int c = 0; c < 8; ++c) {
        float d = S->dist[l16][c0 + c];
        if (d < bmax) {
            bd[bpos] = d; bi[bpos] = m0 + c0 + c;
            bmax = bd[0]; bpos = 0;
#pragma unroll
            for (int i = 1; i < 16; ++i)
                if (bd[i] > bmax) { bmax = bd[i]; bpos = i; }
        }
    }
    __builtin_amdgcn_wave_barrier();
}

__global__ void __launch_bounds__(256) k_knn(const float* __restrict__ pnorm,
                                             const float* __restrict__ sqn,
                                             int* __restrict__ nnidx) {
    __shared__ WaveScratch sm[8];
    const int lane = threadIdx.x & 31;
    const int wv   = threadIdx.x >> 5;
    const int l16  = lane & 15;
    const int lh   = lane >> 4;                 // lane half: 0 or 1
    const int gt   = blockIdx.x * 8 + wv;       // 0 .. 1567
    const int b    = gt / RT;
    const int n0   = (gt % RT) << 4;
    const float* P  = pnorm + (size_t)b * NN * CC;
    const float* SQ = sqn + b * NN;
    WaveScratch* S = &sm[wv];

    // A operand: rows n0..n0+15, fp32 16x4 layout, all 16 K-steps preloaded.
    v2f a[16];
    load_btile(a, P + (size_t)(n0 + l16) * CC, lh);
    float rsq[8];
#pragma unroll
    for (int j = 0; j < 8; ++j) rsq[j] = SQ[n0 + lh * 8 + j];

    float bd[16]; int bi[16];
#pragma unroll
    for (int i = 0; i < 16; ++i) { bd[i] = 3.0e38f; bi[i] = 0; }
    float bmax = 3.0e38f; int bpos = 0;

    // Ping-pong pipeline over the 196 column tiles (even count).
    v2f b0[16], b1[16];
    load_btile(b0, P + (size_t)l16 * CC, lh);
    for (int m0 = 0; m0 < NN; m0 += 32) {
        load_btile(b1, P + (size_t)(m0 + 16 + l16) * CC, lh);       // tile m0+16
        knn_tile(a, b0, m0, SQ, rsq, S, l16, lh, bd, bi, bmax, bpos);
        if (m0 + 32 < NN)
            load_btile(b0, P + (size_t)(m0 + 32 + l16) * CC, lh);   // tile m0+32
        knn_tile(a, b1, m0 + 16, SQ, rsq, S, l16, lh, bd, bi, bmax, bpos);
    }

    // Merge lane L with lane L+16 (two column halves of the same row).
#pragma unroll
    for (int i = 0; i < 16; ++i) {
        S->md[l16][lh * 16 + i] = bd[i];
        S->mi[l16][lh * 16 + i] = bi[i];
    }
    __builtin_amdgcn_wave_barrier();
    if (lane < 16) {
        int* o = nnidx + ((size_t)b * NN + n0 + lane) * KK;
        for (int t = 0; t < KK; ++t) {
            float best = 3.5e38f; int bp = 0;
#pragma unroll
            for (int i = 0; i < 32; ++i) {
                float d = S->md[lane][i];
                if (d < best) { best = d; bp = i; }
            }
            o[t] = S->mi[lane][bp];
            S->md[lane][bp] = 3.5e38f;
        }
    }
}

// ---------------------------------------------------------------------------
// Attention projection folding:  a1 = e.w1 = pts.(W_emb@w1) + b_emb.w1  (assoc)
// k_attvec precomputes u1,u2 (64-vectors) and the two bias constants.
// ---------------------------------------------------------------------------
__global__ void k_attvec(const float* __restrict__ Wemb, const float* __restrict__ bemb,
                         const float* __restrict__ Watt, float* __restrict__ uv) {
    int c = threadIdx.x;                      // 0..63
    if (c < CC) {
        const float* row = Wemb + (size_t)c * CC;
        float s1 = 0.f, s2 = 0.f;
#pragma unroll
        for (int o = 0; o < CC; ++o) { float v = row[o]; s1 += v * Watt[o]; s2 += v * Watt[CC + o]; }
        uv[c]      = s1;
        uv[CC + c] = s2;
    }
    if (c == 0) {
        float c1 = 0.f, c2 = 0.f;
#pragma unroll
        for (int o = 0; o < CC; ++o) { c1 += bemb[o] * Watt[o]; c2 += bemb[o] * Watt[CC + o]; }
        uv[2 * CC]     = c1;
        uv[2 * CC + 1] = c2;
    }
}

// a1[t] = pts[t].u1 + c1 ;  a2[t] = pts[t].u2 + c2
__global__ void k_att_proj(const float* __restrict__ pts, const float* __restrict__ uv,
                           float* __restrict__ a1, float* __restrict__ a2) {
    int t = blockIdx.x * 256 + threadIdx.x;
    if (t >= CNT) return;
    const float4* r  = (const float4*)(pts + (size_t)t * CC);
    const float4* U1 = (const float4*)uv;
    const float4* U2 = (const float4*)(uv + CC);
    float s1 = uv[2 * CC], s2 = uv[2 * CC + 1];
#pragma unroll
    for (int i = 0; i < 16; ++i) {
        float4 v = r[i], w1 = U1[i], w2 = U2[i];
        s1 += v.x*w1.x + v.y*w1.y + v.z*w1.z + v.w*w1.w;
        s2 += v.x*w2.x + v.y*w2.y + v.z*w2.z + v.w*w2.w;
    }
    a1[t] = s1; a2[t] = s2;
}

// ---------------------------------------------------------------------------
// Attention softmax over 16 neighbors + weighted aggregation of raw features.
// ---------------------------------------------------------------------------
__global__ void k_attn_agg(const float* __restrict__ pts, const int* __restrict__ nnidx,
                           const float* __restrict__ a1, const float* __restrict__ a2,
                           const float* __restrict__ batt, float* __restrict__ agg) {
    int t = blockIdx.x * 256 + threadIdx.x;   // b*N + n
    if (t >= CNT) return;
    const int b = t / NN;
    const float* Pb = pts + (size_t)b * NN * CC;
    const float* A2 = a2 + b * NN;
    const int* idx = nnidx + (size_t)t * KK;

    float lg[KK]; int id[KK];
    const float s1 = a1[t] + batt[0];
    float m = -3.5e38f;
#pragma unroll
    for (int k = 0; k < KK; ++k) {
        int j = idx[k]; id[k] = j;
        float l = s1 + A2[j];
        l = (l > 0.f) ? l : 0.1f * l;          // leaky_relu(0.1)
        lg[k] = l; m = fmaxf(m, l);
    }
    float ssum = 0.f;
#pragma unroll
    for (int k = 0; k < KK; ++k) { float w = __expf(lg[k] - m); lg[k] = w; ssum += w; }
    const float inv = 1.f / ssum;

    float4 acc[16];
#pragma unroll
    for (int i = 0; i < 16; ++i) acc[i] = make_float4(0.f, 0.f, 0.f, 0.f);
    for (int k = 0; k < KK; ++k) {
        const float w = lg[k] * inv;
        const float4* nr = (const float4*)(Pb + (size_t)id[k] * CC);
#pragma unroll
        for (int i = 0; i < 16; ++i) {
            float4 v = nr[i];
            acc[i].x += w * v.x; acc[i].y += w * v.y;
            acc[i].z += w * v.z; acc[i].w += w * v.w;
        }
    }
    float4* o = (float4*)(agg + (size_t)t * CC);
#pragma unroll
    for (int i = 0; i < 16; ++i) o[i] = acc[i];
}

// ---------------------------------------------------------------------------
// y = [pts | agg] @ W_conv + b_conv  (K=128 fp32 WMMA) + fused BN stats.
// W_conv is staged transposed into LDS with 132-float pitch so each B
// fragment is one conflict-free ds_load_b64 (banks (4*col + k) mod 64).
// ---------------------------------------------------------------------------
__global__ void __launch_bounds__(256) k_conv(const float* __restrict__ pts,
                                              const float* __restrict__ agg,
                                              const float* __restrict__ Wc,
                                              const float* __restrict__ bcv,
                                              float* __restrict__ y,
                                              float* __restrict__ csum,
                                              float* __restrict__ csq) {
    __shared__ float wt[OUTC * 132];          // wt[col*132 + k] = Wc[k*64 + col]
    for (int i = threadIdx.x; i < 2 * CC * OUTC; i += 256) {
        int k = i >> 6, col = i & 63;
        wt[col * 132 + k] = Wc[i];
    }
    __syncthreads();

    const int lane = threadIdx.x & 31, wv = threadIdx.x >> 5;
    const int l16 = lane & 15, lh = lane >> 4;
    const int gt = blockIdx.x * 8 + wv;
    const int b = gt / RT, n0 = (gt % RT) << 4;

    const float* ar1 = pts + ((size_t)b * NN + n0 + l16) * CC;
    const float* ar2 = agg + ((size_t)b * NN + n0 + l16) * CC;
    v2f a[32];
#pragma unroll
    for (int s = 0; s < 16; ++s) {
        int k0 = 4 * s + (lh << 1);
        a[s][0]      = ar1[k0]; a[s][1]      = ar1[k0 + 1];
        a[16 + s][0] = ar2[k0]; a[16 + s][1] = ar2[k0 + 1];
    }
    float* Y = y + ((size_t)b * NN + n0) * OUTC;
#pragma unroll
    for (int ct = 0; ct < 4; ++ct) {
        const int col = ct * 16 + l16;
        const float* wr = wt + (size_t)col * 132;
        v2f bw[32];                            // batched ds_load_b64
#pragma unroll
        for (int s = 0; s < 32; ++s) {
            int k = 4 * (s & 15) + (lh << 1) + ((s >> 4) << 6);
            bw[s][0] = wr[k]; bw[s][1] = wr[k + 1];
        }
        v8f acc = {0.f, 0.f, 0.f, 0.f, 0.f, 0.f, 0.f, 0.f};
#pragma unroll
        for (int s = 0; s < 32; ++s) acc = wmma_f32_k4(a[s], bw[s], acc);

        const float bias = bcv[col];
        float s1 = 0.f, s2 = 0.f;
#pragma unroll
        for (int j = 0; j < 8; ++j) {
            float v = acc[j] + bias;
            Y[(size_t)(lh * 8 + j) * OUTC + col] = v;
            s1 += v; s2 += v * v;
        }
        atomicAdd(&csum[col], s1);
        atomicAdd(&csq[col],  s2);
    }
}

// ---------------------------------------------------------------------------
// BN finalize + ReLU + residual + transpose store to [B, C, H, W].
// ---------------------------------------------------------------------------
__global__ void k_final(const float* __restrict__ y, const float* __restrict__ pts,
                        const float* __restrict__ csum, const float* __restrict__ csq,
                        const float* __restrict__ gamma, const float* __restrict__ beta,
                        float* __restrict__ out) {
    int t = blockIdx.x * 256 + threadIdx.x;   // (b*N+n)*C + c
    if (t >= BB * NN * OUTC) return;
    int c = t & 63;
    int bn = t >> 6;
    int n = bn % NN, b = bn / NN;
    const float cnt = (float)CNT;
    float mu  = csum[c] / cnt;
    float var = csq[c] / cnt - mu * mu;       // biased, matches torch BN training
    float v = (y[t] - mu) * rsqrtf(var + BN_EPS) * gamma[c] + beta[c];
    v = fmaxf(v, 0.f);
    v += pts[t];
    out[((size_t)b * OUTC + c) * NN + n] = v;
}

// ---------------------------------------------------------------------------
// Host launcher. Workspace usage: ~27 MB (4x [B,N,C] f32 buffers + indices).
// ---------------------------------------------------------------------------
extern "C" void kernel_launch(void* const* d_in, const int* in_sizes, int n_in,
                              void* d_out, int out_size, void* d_ws, size_t ws_size,
                              hipStream_t stream) {
    const float* x     = (const float*)d_in[0];
    const float* Wemb  = (const float*)d_in[1];
    const float* bemb  = (const float*)d_in[2];
    const float* Watt  = (const float*)d_in[3];
    const float* batt  = (const float*)d_in[4];
    const float* Wc    = (const float*)d_in[5];
    const float* bcv   = (const float*)d_in[6];
    const float* gamma = (const float*)d_in[7];
    const float* beta  = (const float*)d_in[8];
    float* out = (float*)d_out;
    (void)in_sizes; (void)n_in; (void)out_size; (void)ws_size;

    char* ws = (char*)d_ws;
    size_t off = 0;
    auto alloc = [&](size_t bytes) -> char* {
        char* p = ws + off;
        off = (off + bytes + 255) & ~(size_t)255;
        return p;
    };
    float* pts   = (float*)alloc(sizeof(float) * BB * NN * CC);
    float* pnorm = (float*)alloc(sizeof(float) * BB * NN * CC);
    float* cn    = (float*)alloc(sizeof(float) * BB * CC);
    float* sqn   = (float*)alloc(sizeof(float) * CNT);
    int*   nn    = (int*)  alloc(sizeof(int)   * CNT * KK);
    float* uv    = (float*)alloc(sizeof(float) * (2 * CC + 2));
    float* a1    = (float*)alloc(sizeof(float) * CNT);
    float* a2    = (float*)alloc(sizeof(float) * CNT);
    float* agg   = (float*)alloc(sizeof(float) * BB * NN * CC);
    float* yb    = (float*)alloc(sizeof(float) * CNT * OUTC);
    float* cst   = (float*)alloc(sizeof(float) * 2 * OUTC);
    float* csum  = cst;
    float* csq   = cst + OUTC;

    k_zero     <<<1, 2 * OUTC, 0, stream>>>(cst, 2 * OUTC);
    k_colnorm  <<<BB * CC, 256, 0, stream>>>(x, cn);
    k_transpose<<<(BB * NN * CC + 255) / 256, 256, 0, stream>>>(x, cn, pts, pnorm);
    k_sqnorm   <<<(CNT + 255) / 256, 256, 0, stream>>>(pnorm, sqn);
    k_knn      <<<RT, 256, 0, stream>>>(pnorm, sqn, nn);      // 196 blocks = 1568 waves
    k_attvec   <<<1, CC, 0, stream>>>(Wemb, bemb, Watt, uv);
    k_att_proj <<<(CNT + 255) / 256, 256, 0, stream>>>(pts, uv, a1, a2);
    k_attn_agg <<<(CNT + 255) / 256, 256, 0, stream>>>(pts, nn, a1, a2, batt, agg);
    k_conv     <<<RT, 256, 0, stream>>>(pts, agg, Wc, bcv, yb, csum, csq);
    k_final    <<<(BB * NN * OUTC + 255) / 256, 256, 0, stream>>>(yb, pts, csum, csq, gamma, beta, out);
}